// PrivateMultihopAggregation_27986006901090
// MI455X (gfx1250) — compile-verified
//
#include <hip/hip_runtime.h>
#include <stdint.h>

typedef __attribute__((ext_vector_type(2))) float v2f;
typedef __attribute__((ext_vector_type(8))) float v8f;

#define NOISE 0.1f
#define EPS 1e-12f

// ---------------- utility: deterministic gaussian noise ----------------
__device__ __forceinline__ uint32_t hash32(uint32_t x) {
    x ^= x >> 16; x *= 0x7feb352dU;
    x ^= x >> 15; x *= 0x846ca68bU;
    x ^= x >> 16;
    return x;
}

__device__ __forceinline__ float2 gauss2(uint32_t seed) {
    uint32_t h1 = hash32(seed);
    uint32_t h2 = hash32(seed ^ 0x9e3779b9U);
    float u1 = (float)(h1 >> 8) * (1.0f / 16777216.0f) + 1e-7f;
    float u2 = (float)(h2 >> 8) * (1.0f / 16777216.0f);
    float r = sqrtf(-2.0f * __logf(u1));
    float t = 6.28318530718f * u2;
    float2 g; g.x = r * __cosf(t); g.y = r * __sinf(t);
    return g;
}

// ---------------- CSR construction ----------------
__global__ void zero_i32(int* __restrict__ p, int n) {
    int i = blockIdx.x * blockDim.x + threadIdx.x;
    if (i < n) p[i] = 0;
}

__global__ void degree_kernel(const int* __restrict__ row, int* __restrict__ deg, int E) {
    int i = blockIdx.x * blockDim.x + threadIdx.x;
    if (i < E) atomicAdd(&deg[row[i]], 1);
}

// Single-workgroup exclusive scan over n bins (n/1024 sequential chunks).
// Also resets deg[] to zero so it can be reused as the fill cursor.
__global__ __launch_bounds__(1024) void scan_kernel(int* __restrict__ deg,
                                                    int* __restrict__ rowptr, int n) {
    __shared__ int sh[1024];
    __shared__ int carry;
    int tid = threadIdx.x;
    if (tid == 0) carry = 0;
    __syncthreads();
    for (int base = 0; base < n; base += 1024) {
        int i = base + tid;
        int v = (i < n) ? deg[i] : 0;
        if (i < n) deg[i] = 0;          // reset for fill phase
        sh[tid] = v;
        __syncthreads();
        for (int off = 1; off < 1024; off <<= 1) {
            int t = (tid >= off) ? sh[tid - off] : 0;
            __syncthreads();
            sh[tid] += t;
            __syncthreads();
        }
        int c = carry;
        if (i < n) rowptr[i] = c + sh[tid] - v;   // exclusive
        __syncthreads();
        if (tid == 0) carry = c + sh[1023];
        __syncthreads();
    }
    if (tid == 0) rowptr[n] = carry;
}

__global__ void fill_kernel(const int* __restrict__ row, const int* __restrict__ col,
                            const int* __restrict__ rowptr, int* __restrict__ cursor,
                            int* __restrict__ colS, int E) {
    int i = blockIdx.x * blockDim.x + threadIdx.x;
    if (i < E) {
        int r = row[i];
        int p = atomicAdd(&cursor[r], 1);
        colS[rowptr[r] + p] = col[i];
    }
}

// ---------------- out[0] = l2_normalize(x), via V_WMMA_F32_16X16X4_F32 ----------------
// One wave handles 16 rows. Gram diagonal = per-row sum of squares:
// G = sum_k A_k * A_k^T where A_k is the 16x4 chunk of the row tile.
// In the CDNA5 wave32 layout the A (16x4, MxK) and B (4x16, KxN with N==M) operands
// hold identical per-lane data, so a single float2 load feeds both.
__global__ __launch_bounds__(256) void normalize0_wmma(const float* __restrict__ x,
                                                       float* __restrict__ out, int n) {
    int lane = threadIdx.x & 31;
    int tile = blockIdx.x * 8 + (threadIdx.x >> 5);   // wave-uniform
    long base = (long)tile * 16;
    if (base >= n) return;

    if (base + 16 <= (long)n) {
        int m = lane & 15;
        int koff = (lane < 16) ? 0 : 2;
        const float* rowp = x + (base + m) * 64 + koff;

        v8f c = {};
#pragma unroll
        for (int kc = 0; kc < 64; kc += 4) {
            v2f a;
            a.x = rowp[kc];
            a.y = rowp[kc + 1];
            // D = A(16x4) * B(4x16) + C ; B == A^T has identical lane data
            c = __builtin_amdgcn_wmma_f32_16x16x4_f32(
                    false, a, false, a, (short)0, c, false, false);
        }

        // diag(M=N): rows 0..7 live in lane M, vgpr M; rows 8..15 in lane 24+v, vgpr v
        int idx = (lane < 8) ? lane : ((lane >= 24) ? (lane - 24) : 0);
        float ss = c[0];
        if (idx == 1) ss = c[1];
        if (idx == 2) ss = c[2];
        if (idx == 3) ss = c[3];
        if (idx == 4) ss = c[4];
        if (idx == 5) ss = c[5];
        if (idx == 6) ss = c[6];
        if (idx == 7) ss = c[7];
        float inv = 1.0f / fmaxf(sqrtf(ss), EPS);

        const v2f* xin = (const v2f*)(x + base * 64);
        v2f* o = (v2f*)(out + base * 64);
#pragma unroll
        for (int mm = 0; mm < 16; ++mm) {
            float s = __shfl(inv, (mm < 8) ? mm : (mm + 16), 32);
            v2f v = xin[mm * 32 + lane];
            v.x *= s; v.y *= s;
            o[mm * 32 + lane] = v;
        }
    } else {
        // tail tile: plain wave-per-row normalization
        for (long r = base; r < n; ++r) {
            v2f v = ((const v2f*)(x + r * 64))[lane];
            float ss = v.x * v.x + v.y * v.y;
            ss += __shfl_xor(ss, 1, 32);
            ss += __shfl_xor(ss, 2, 32);
            ss += __shfl_xor(ss, 4, 32);
            ss += __shfl_xor(ss, 8, 32);
            ss += __shfl_xor(ss, 16, 32);
            float s = 1.0f / fmaxf(sqrtf(ss), EPS);
            v.x *= s; v.y *= s;
            ((v2f*)(out + r * 64))[lane] = v;
        }
    }
}

// ---------------- fused hop: CSR gather-sum + noise + normalize ----------------
// One wave32 per destination row. Edge indices loaded 32-at-a-time (coalesced),
// broadcast via shfl; source rows gathered as two float4 half-waves (b128 loads).
__global__ __launch_bounds__(256) void hop_kernel(const float* __restrict__ in,
                                                  float* __restrict__ out,
                                                  const int* __restrict__ rowptr,
                                                  const int* __restrict__ cols,
                                                  int n, int hop) {
    int lane = threadIdx.x & 31;
    int r = blockIdx.x * 8 + (threadIdx.x >> 5);      // wave-uniform
    if (r >= n) return;

    int s = rowptr[r];
    int e = rowptr[r + 1];
    int half = lane >> 4;       // 0: even edges, 1: odd edges
    int lcol = lane & 15;       // owns columns 4*lcol .. 4*lcol+3

    float ax = 0.f, ay = 0.f, az = 0.f, aw = 0.f;

    for (int i = s; i < e; i += 32) {
        if (i + 32 < e) __builtin_prefetch(&cols[i + 32 + lane], 0, 0); // global_prefetch_b8
        int cidx = (i + lane < e) ? cols[i + lane] : 0;
        int cnt = e - i; if (cnt > 32) cnt = 32;
        for (int j = half; j < cnt; j += 2) {
            int cc = __shfl(cidx, j, 32);
            const float4* src = (const float4*)(in + (size_t)cc * 64);
            float4 v = src[lcol];
            ax += v.x; ay += v.y; az += v.z; aw += v.w;
        }
    }

    // combine even/odd half-wave partial sums
    ax += __shfl_xor(ax, 16, 32);
    ay += __shfl_xor(ay, 16, 32);
    az += __shfl_xor(az, 16, 32);
    aw += __shfl_xor(aw, 16, 32);

    // additive gaussian noise (deterministic per (hop,row,col))
    uint32_t seed = ((uint32_t)hop << 28) ^ ((uint32_t)r * 2654435761u) ^ ((uint32_t)lcol * 40503u);
    float2 g0 = gauss2(seed);
    float2 g1 = gauss2(seed ^ 0x85ebca6bU);
    ax += NOISE * g0.x; ay += NOISE * g0.y;
    az += NOISE * g1.x; aw += NOISE * g1.y;

    // row L2 norm across the 16 column-group lanes (masks < 16 keep halves separate;
    // both halves hold identical data after the xor-16 combine)
    float ss = ax * ax + ay * ay + az * az + aw * aw;
    ss += __shfl_xor(ss, 1, 32);
    ss += __shfl_xor(ss, 2, 32);
    ss += __shfl_xor(ss, 4, 32);
    ss += __shfl_xor(ss, 8, 32);
    float inv = 1.0f / fmaxf(sqrtf(ss), EPS);

    if (half == 0) {
        float4 o;
        o.x = ax * inv; o.y = ay * inv; o.z = az * inv; o.w = aw * inv;
        ((float4*)(out + (size_t)r * 64))[lcol] = o;
    }
}

// ---------------- host launcher ----------------
extern "C" void kernel_launch(void* const* d_in, const int* in_sizes, int n_in,
                              void* d_out, int out_size, void* d_ws, size_t ws_size,
                              hipStream_t stream) {
    const float* x = (const float*)d_in[0];
    const int* ei = (const int*)d_in[1];
    int n = in_sizes[0] / 64;       // 100000
    int E = in_sizes[1] / 2;        // 1600000
    const int* row = ei;            // edge_index[0]
    const int* col = ei + E;        // edge_index[1]
    float* out = (float*)d_out;     // [4, n, 64]

    int* rowptr = (int*)d_ws;       // n+1
    int* cursor = rowptr + (n + 1); // n   (degree, then fill cursor)
    int* colS = cursor + n;         // E   (column indices sorted by dst row)

    zero_i32<<<(n + 255) / 256, 256, 0, stream>>>(cursor, n);
    degree_kernel<<<(E + 255) / 256, 256, 0, stream>>>(row, cursor, E);
    scan_kernel<<<1, 1024, 0, stream>>>(cursor, rowptr, n);
    fill_kernel<<<(E + 255) / 256, 256, 0, stream>>>(row, col, rowptr, cursor, colS, E);

    int ntiles = (n + 15) / 16;
    normalize0_wmma<<<(ntiles + 7) / 8, 256, 0, stream>>>(x, out, n);

    for (int k = 0; k < 3; ++k) {
        const float* src = out + (size_t)k * n * 64;
        float* dst = out + (size_t)(k + 1) * n * 64;
        hop_kernel<<<(n + 7) / 8, 256, 0, stream>>>(src, dst, rowptr, colS, n, k);
    }
}